// GCNNet_867583394113
// MI455X (gfx1250) — compile-verified
//
#include <hip/hip_runtime.h>
#include <stdint.h>

// ---------------------------------------------------------------------------
// GCN (3x GCNConv, D=128) for MI455X / gfx1250.
//   lin  = act(h) @ W            -- V_WMMA_F32_16X16X4_F32 (fp32-exact GEMM)
//   agg  = b + dinv^2 * lin      -- self-loop + bias
//   agg += scatter(dinv_s*dinv_d * lin[src] -> dst)  -- global_atomic_add_f32
// ---------------------------------------------------------------------------

typedef __attribute__((ext_vector_type(2))) float v2f;
typedef __attribute__((ext_vector_type(8))) float v8f;

#define DIM 128

// native fp32 global atomic add (no return -> STOREcnt, L2-side RMW)
__device__ __forceinline__ void gfadd(float* p, float v) {
    asm volatile("global_atomic_add_f32 %0, %1, off" :: "v"(p), "v"(v) : "memory");
}

// ---------------------------------------------------------------------------
// degree / norm kernels
// ---------------------------------------------------------------------------
__global__ void k_deg_init(float* __restrict__ deg, int n) {
    int i = blockIdx.x * blockDim.x + threadIdx.x;
    if (i < n) deg[i] = 1.0f;                      // self-loop
}

__global__ void k_deg_accum(const int* __restrict__ dst, float* __restrict__ deg, int e) {
    int i = blockIdx.x * blockDim.x + threadIdx.x;
    if (i < e) gfadd(&deg[dst[i]], 1.0f);
}

__global__ void k_rsqrt(float* __restrict__ d, int n) {
    int i = blockIdx.x * blockDim.x + threadIdx.x;
    if (i < n) d[i] = rsqrtf(d[i]);
}

// ---------------------------------------------------------------------------
// GEMM: out[N x 128] = act(A[N x 128]) @ W[128 x 128]
// One block = 8 waves; block owns a 16-row M-tile, wave w owns N-cols [16w,16w+16).
// A tile staged in LDS (padded stride 132 words -> conflict-free b64 reads).
// 32 chained V_WMMA_F32_16X16X4_F32 per wave over K=128.
// ---------------------------------------------------------------------------
__global__ __launch_bounds__(256)
void k_gemm_wmma(const float* __restrict__ A, const float* __restrict__ W,
                 float* __restrict__ out, int n, int relu_in) {
    __shared__ float a_lds[16 * 132];

    const int m0 = blockIdx.x * 16;
    const int t  = threadIdx.x;

    // stage 16x128 A tile (float4, coalesced), fold relu of previous layer
    #pragma unroll
    for (int c = t; c < 512; c += 256) {           // 512 float4 chunks
        int r    = c >> 5;                         // 32 float4 per row
        int col4 = (c & 31) << 2;
        int row  = m0 + r;
        float4 v = make_float4(0.f, 0.f, 0.f, 0.f);
        if (row < n) v = *(const float4*)(A + (size_t)row * DIM + col4);
        if (relu_in) {
            v.x = fmaxf(v.x, 0.f); v.y = fmaxf(v.y, 0.f);
            v.z = fmaxf(v.z, 0.f); v.w = fmaxf(v.w, 0.f);
        }
        *(float4*)(&a_lds[r * 132 + col4]) = v;
    }
    __syncthreads();

    const int wave = t >> 5;
    const int lane = t & 31;
    const int half = lane >> 4;                    // 0: K+0/1, 1: K+2/3 (A); rows for B
    const int l    = lane & 15;
    const int n0   = wave * 16;

    v8f acc = {};
    #pragma unroll
    for (int k = 0; k < DIM; k += 4) {
        // A fragment: 16x4, lane l row, element = K (+2 per lane-half)
        v2f a = *(const v2f*)(&a_lds[l * 132 + k + 2 * half]);
        // B fragment: 4x16, row K+2*half (+1 in second element), col n0+l
        const float* bp = W + (size_t)(k + 2 * half) * DIM + n0 + l;
        v2f b;
        b.x = bp[0];
        b.y = bp[DIM];
        acc = __builtin_amdgcn_wmma_f32_16x16x4_f32(
                  false, a, false, b, (short)0, acc, false, false);
    }

    // C/D layout: VGPR r -> M = m0 + r + 8*half, N = n0 + l
    #pragma unroll
    for (int r = 0; r < 8; ++r) {
        int row = m0 + r + 8 * half;
        if (row < n) out[(size_t)row * DIM + n0 + l] = acc[r];
    }
}

// ---------------------------------------------------------------------------
// agg[i,:] = bias + dinv[i]^2 * lin[i,:]   (self-loop contribution + bias)
// one thread = one float4 chunk
// ---------------------------------------------------------------------------
__global__ void k_agg_init(const float* __restrict__ lin, const float* __restrict__ dinv,
                           const float* __restrict__ bias, float* __restrict__ agg, int n) {
    int i = blockIdx.x * blockDim.x + threadIdx.x;
    if (i >= n * 32) return;
    int row = i >> 5;
    int c4  = (i & 31) << 2;
    float di = dinv[row];
    float s  = di * di;
    float4 v = *(const float4*)(lin  + (size_t)row * DIM + c4);
    float4 b = *(const float4*)(bias + c4);
    float4 o = make_float4(b.x + s * v.x, b.y + s * v.y,
                           b.z + s * v.z, b.w + s * v.w);
    *(float4*)(agg + (size_t)row * DIM + c4) = o;
}

// ---------------------------------------------------------------------------
// edge scatter: agg[dst,:] += dinv[src]*dinv[dst] * lin[src,:]
// one wave per edge, lane = float4 chunk, 4x global_atomic_add_f32 per lane
// ---------------------------------------------------------------------------
__global__ void k_edge_scatter(const float* __restrict__ lin, const float* __restrict__ dinv,
                               const int* __restrict__ src, const int* __restrict__ dst,
                               float* __restrict__ agg, int e) {
    int i = blockIdx.x * blockDim.x + threadIdx.x;
    int edge = i >> 5;
    if (edge >= e) return;
    int lane = i & 31;
    int s = src[edge];
    int d = dst[edge];
    float c = dinv[s] * dinv[d];
    int c4 = lane << 2;
    float4 v = *(const float4*)(lin + (size_t)s * DIM + c4);
    float* p = agg + (size_t)d * DIM + c4;
    gfadd(p + 0, c * v.x);
    gfadd(p + 1, c * v.y);
    gfadd(p + 2, c * v.z);
    gfadd(p + 3, c * v.w);
}

// ---------------------------------------------------------------------------
// orchestration
// ---------------------------------------------------------------------------
extern "C" void kernel_launch(void* const* d_in, const int* in_sizes, int n_in,
                              void* d_out, int out_size, void* d_ws, size_t ws_size,
                              hipStream_t stream) {
    (void)n_in; (void)out_size; (void)ws_size;

    const float* x  = (const float*)d_in[0];
    const int*   ei = (const int*)  d_in[1];     // [2, E] flat: row0 = src, row1 = dst
    const float* W1 = (const float*)d_in[2];
    const float* b1 = (const float*)d_in[3];
    const float* W2 = (const float*)d_in[4];
    const float* b2 = (const float*)d_in[5];
    const float* W3 = (const float*)d_in[6];
    const float* b3 = (const float*)d_in[7];

    const int N = in_sizes[0] / DIM;
    const int E = in_sizes[1] / 2;

    float* out  = (float*)d_out;                 // doubles as agg buffer
    float* dinv = (float*)d_ws;
    float* lin  = dinv + (((size_t)N + 255) & ~(size_t)255);   // N*128 floats

    const int* src = ei;
    const int* dst = ei + E;

    const int B = 256;
    const int gN   = (N + B - 1) / B;
    const int gE   = (E + B - 1) / B;
    const int gNf  = (N * 32 + B - 1) / B;       // float4 chunks over N*128
    const int gEf  = (int)(((long long)E * 32 + B - 1) / B);
    const int gMt  = (N + 15) / 16;              // WMMA M-tiles

    // normalization: deg (with self-loop) -> dinv = rsqrt(deg)
    k_deg_init <<<gN, B, 0, stream>>>(dinv, N);
    k_deg_accum<<<gE, B, 0, stream>>>(dst, dinv, E);
    k_rsqrt    <<<gN, B, 0, stream>>>(dinv, N);

    // layer 1: lin = x @ W1 ; out = b1 + selfloop + scatter
    k_gemm_wmma   <<<gMt, 256, 0, stream>>>(x, W1, lin, N, 0);
    k_agg_init    <<<gNf, B,   0, stream>>>(lin, dinv, b1, out, N);
    k_edge_scatter<<<gEf, B,   0, stream>>>(lin, dinv, src, dst, out, E);

    // layer 2: lin = relu(out) @ W2 ; out = b2 + selfloop + scatter
    k_gemm_wmma   <<<gMt, 256, 0, stream>>>(out, W2, lin, N, 1);
    k_agg_init    <<<gNf, B,   0, stream>>>(lin, dinv, b2, out, N);
    k_edge_scatter<<<gEf, B,   0, stream>>>(lin, dinv, src, dst, out, E);

    // layer 3: lin = relu(out) @ W3 ; out = b3 + selfloop + scatter (no relu)
    k_gemm_wmma   <<<gMt, 256, 0, stream>>>(out, W3, lin, N, 1);
    k_agg_init    <<<gNf, B,   0, stream>>>(lin, dinv, b3, out, N);
    k_edge_scatter<<<gEf, B,   0, stream>>>(lin, dinv, src, dst, out, E);
}